// ReservoirSampler_65884798320942
// MI455X (gfx1250) — compile-verified
//
#include <hip/hip_runtime.h>
#include <hip/hip_bf16.h>

#define RS_D 256  // feature dim, fixed by the module

#ifndef __has_builtin
#define __has_builtin(x) 0
#endif

#if __has_builtin(__builtin_amdgcn_global_load_async_to_lds_b128) && \
    __has_builtin(__builtin_amdgcn_global_store_async_from_lds_b128) && \
    __has_builtin(__builtin_amdgcn_s_wait_asynccnt)
#define RS_ASYNC 1
#else
#define RS_ASYNC 0
#endif

typedef float __attribute__((ext_vector_type(4))) f32x4;
typedef int   __attribute__((ext_vector_type(4))) v4i;
typedef __attribute__((address_space(1))) v4i* gptr_v4i;  // global int4*
typedef __attribute__((address_space(3))) v4i* lptr_v4i;  // LDS int4*

// ---------------------------------------------------------------------------
// Phase 2 (defined first so the disasm snippet shows the async path):
// per-slot select + row copy. 64 lanes x 16B = one 1KB row; 4 rows per
// 256-thread block. Bulk data moves over the CDNA5 async global->LDS->global
// path (ASYNCcnt), b128 per lane.
// ---------------------------------------------------------------------------
__global__ void rs_copy_kernel(const float* __restrict__ samples,
                               const float* __restrict__ buffer,
                               const int* __restrict__ last_p,
                               float* __restrict__ out, int n) {
  __shared__ float tile[4 * RS_D];
  const int tid = threadIdx.x;
  const int row = blockIdx.x * 4 + (tid >> 6);
  const int c   = (tid & 63) << 2;          // float column offset (16B granule)
  if (row >= n) return;

  const int lp = last_p[row];
  const float* src = (lp >= 0) ? (samples + (size_t)lp * RS_D)
                               : (buffer + (size_t)row * RS_D);
  float* dst = out + (size_t)row * RS_D + c;

#if RS_ASYNC
  // global -> LDS (async, b128), per-lane addresses
  __builtin_amdgcn_global_load_async_to_lds_b128(
      (gptr_v4i)(src + c), (lptr_v4i)&tile[tid * 4], 0, 0);
  // own-lane LDS RAW: drain ASYNCcnt before sourcing the store from LDS
  __builtin_amdgcn_s_wait_asynccnt(0);
  // LDS -> global (async, b128); S_ENDPGM's implicit wait-idle covers completion
  __builtin_amdgcn_global_store_async_from_lds_b128(
      (gptr_v4i)dst, (lptr_v4i)&tile[tid * 4], 0, 0);
#else
  *(f32x4*)dst = *(const f32x4*)(src + c);
#endif
}

// ---------------------------------------------------------------------------
// Phase 1: last-write-wins winner resolution via atomicMax.
// Rejected samples (idx >= n) issue no atomic at all.
// ---------------------------------------------------------------------------
__global__ void rs_scatter_kernel(const float* __restrict__ u,
                                  const int* __restrict__ i0_ptr,
                                  int* __restrict__ last_p,
                                  int b, int n) {
  int p = blockIdx.x * blockDim.x + threadIdx.x;
  if (p >= b) return;
  int i0  = *i0_ptr;          // device scalar (single-element input array)
  int pos = i0 + p;           // counter value seen by sample p
  int idx;
  if (pos < n) {
    idx = pos;                // deterministic fill phase
  } else {
    // randint(0, pos) inclusive == floor(u * (pos+1)), clipped for u -> 1
    float f = u[p] * (float)(pos + 1);
    int r = (int)floorf(f);
    r = min(r, pos);
    idx = r;
  }
  if (idx < n) atomicMax(&last_p[idx], p);
}

// ---------------------------------------------------------------------------
// Phase 0: init winner table to -1 (d_ws is poisoned, must init every call)
// ---------------------------------------------------------------------------
__global__ void rs_init_kernel(int* __restrict__ last_p, int n) {
  int i = blockIdx.x * blockDim.x + threadIdx.x;
  if (i < n) last_p[i] = -1;
}

// ---------------------------------------------------------------------------
// Launch
// ---------------------------------------------------------------------------
extern "C" void kernel_launch(void* const* d_in, const int* in_sizes, int n_in,
                              void* d_out, int out_size, void* d_ws, size_t ws_size,
                              hipStream_t stream) {
  const float* samples = (const float*)d_in[0];   // [B, 256] f32
  const float* buffer  = (const float*)d_in[1];   // [N, 256] f32
  const float* u       = (const float*)d_in[2];   // [B] f32
  const int*   i0_ptr  = (const int*)d_in[3];     // scalar on device
  float* out = (float*)d_out;                      // [N, 256] f32

  const int b = in_sizes[0] / RS_D;
  const int n = in_sizes[1] / RS_D;

  int* last_p = (int*)d_ws;                        // n ints of scratch

  rs_init_kernel<<<(n + 255) / 256, 256, 0, stream>>>(last_p, n);
  rs_scatter_kernel<<<(b + 255) / 256, 256, 0, stream>>>(u, i0_ptr, last_p, b, n);
  rs_copy_kernel<<<(n + 3) / 4, 256, 0, stream>>>(samples, buffer, last_p, out, n);
}